// GATGraphSimilarity_11553462026425
// MI455X (gfx1250) — compile-verified
//
#include <hip/hip_runtime.h>
#include <math.h>

// ---------------- problem constants (match reference) ----------------
#define NNODES  50000
#define NEDGES  800000
#define NGRAPH  64
#define INDIM   128
#define HID     64
#define HEADS1  4
#define F1      (HEADS1*HID)          // 256
#define EPRIME  (NEDGES + NNODES)     // 850000 edges incl. self loops

typedef float v2f __attribute__((ext_vector_type(2)));
typedef float v8f __attribute__((ext_vector_type(8)));

// =====================================================================
// fp32 WMMA GEMM with async-LDS staged B:
//   out[M x NOUT] = A[M x K] @ W[K x NOUT]
// Block = 256 threads (8 waves). The block's K x 64 column-slab of W is
// DMA'd into LDS once via GLOBAL_LOAD_ASYNC_TO_LDS_B128 (ASYNCcnt), then
// each wave computes a 32 x 64 strip (2 M-tiles x 4 N-tiles) with
// V_WMMA_F32_16X16X4_F32, B fragments served from LDS.
// A-frag layout (ISA 7.12.2): lanes 0-15 -> K={k,k+1}, lanes 16-31 ->
// K={k+2,k+3}, all lanes cover M=l16. C/D: VGPR r -> M=r+8*half, N=l16.
// =====================================================================
template<int K, int NOUT>
__global__ void __launch_bounds__(256)
gemm_wmma_lds(const float* __restrict__ A, const float* __restrict__ W,
              float* __restrict__ out)
{
    __shared__ __align__(16) float shW[K * 64];

    const int tid   = threadIdx.x;
    const int nBase = blockIdx.y * 64;

    // ---- async DMA of W column-slab [K x 64] into LDS ----
    {
        const unsigned ldsBase = (unsigned)(size_t)&shW[0];
        #pragma unroll 4
        for (int i = tid; i < K * 16; i += 256) {      // float4 granularity
            const int r  = i >> 4;                     // W row
            const int c4 = i & 15;                     // float4 column
            const unsigned lds  = ldsBase + (unsigned)i * 16u;
            const float*   gptr = W + (size_t)r * NOUT + nBase + c4 * 4;
            asm volatile("global_load_async_to_lds_b128 %0, %1, off"
                         :: "v"(lds), "v"(gptr) : "memory");
        }
        asm volatile("s_wait_asynccnt 0x0" ::: "memory");
    }
    __syncthreads();

    const int wave  = tid >> 5;
    const int lane  = tid & 31;
    const int half  = lane >> 4;       // 0: K pair {0,1}, 1: K pair {2,3}
    const int l16   = lane & 15;
    const int mBase = blockIdx.x * 256 + wave * 32;

    const bool valid0 = (mBase      < NNODES);
    const bool valid1 = (mBase + 16 < NNODES);
    const int  r0 = valid0 ? (mBase      + l16) : 0;   // clamp OOB loads
    const int  r1 = valid1 ? (mBase + 16 + l16) : 0;
    const float* a0p = A + (size_t)r0 * K;
    const float* a1p = A + (size_t)r1 * K;

    v8f acc[2][4] = {};

    for (int k = 0; k < K; k += 4) {
        const int kk = k + 2 * half;
        v2f a0 = *(const v2f*)(a0p + kk);
        v2f a1 = *(const v2f*)(a1p + kk);
        #pragma unroll
        for (int t = 0; t < 4; ++t) {
            const int cl = t * 16 + l16;
            v2f b;
            b.x = shW[kk       * 64 + cl];
            b.y = shW[(kk + 1) * 64 + cl];
            acc[0][t] = __builtin_amdgcn_wmma_f32_16x16x4_f32(
                false, a0, false, b, (short)0, acc[0][t], false, false);
            acc[1][t] = __builtin_amdgcn_wmma_f32_16x16x4_f32(
                false, a1, false, b, (short)0, acc[1][t], false, false);
        }
    }

    #pragma unroll
    for (int mt = 0; mt < 2; ++mt) {
        if (!(mt ? valid1 : valid0)) continue;
        const int rowBase = mBase + mt * 16 + half * 8;
        #pragma unroll
        for (int t = 0; t < 4; ++t) {
            const int col = nBase + t * 16 + l16;
            #pragma unroll
            for (int r = 0; r < 8; ++r)
                out[(size_t)(rowBase + r) * NOUT + col] = acc[mt][t][r];
        }
    }
}

// =====================================================================
// Attention coefficients: as[n,h] = <H[n,h,:], a_src[h,:]>, same for ad.
// =====================================================================
template<int HEADS>
__global__ void attn_coef(const float* __restrict__ Hf,
                          const float* __restrict__ a_src,
                          const float* __restrict__ a_dst,
                          float* __restrict__ as, float* __restrict__ ad)
{
    int t = blockIdx.x * blockDim.x + threadIdx.x;
    if (t >= NNODES * HEADS) return;
    const int n = t / HEADS, h = t % HEADS;
    const float* hp  = Hf + (size_t)n * (HEADS * HID) + h * HID;
    const float* asv = a_src + h * HID;
    const float* adv = a_dst + h * HID;
    float s0 = 0.f, s1 = 0.f;
    #pragma unroll 8
    for (int d = 0; d < HID; ++d) { float v = hp[d]; s0 += v * asv[d]; s1 += v * adv[d]; }
    as[t] = s0; ad[t] = s1;
}

// monotone float <-> uint key (for atomic segment-max)
__device__ __forceinline__ unsigned fkey(float f) {
    unsigned b = __float_as_uint(f);
    return (b & 0x80000000u) ? ~b : (b | 0x80000000u);
}
__device__ __forceinline__ float funkey(unsigned k) {
    return __uint_as_float((k & 0x80000000u) ? (k & 0x7FFFFFFFu) : ~k);
}

// edge src/dst with implicit self loops appended at e >= NEDGES
__device__ __forceinline__ int esrc(const int* ei, int e) {
    return (e < NEDGES) ? ei[e] : (e - NEDGES);
}
__device__ __forceinline__ int edst(const int* ei, int e) {
    return (e < NEDGES) ? ei[NEDGES + e] : (e - NEDGES);
}

// Pass A: leaky-relu edge score, store, atomic segment max over dst
template<int HEADS>
__global__ void edge_scores(const int* __restrict__ ei,
                            const float* __restrict__ as, const float* __restrict__ ad,
                            float* __restrict__ ebuf, unsigned* __restrict__ emax)
{
    int t = blockIdx.x * blockDim.x + threadIdx.x;
    if (t >= EPRIME * HEADS) return;
    const int e = t / HEADS, h = t % HEADS;
    const int s = esrc(ei, e), d = edst(ei, e);
    float x = as[s * HEADS + h] + ad[d * HEADS + h];
    x = (x > 0.f) ? x : 0.2f * x;                  // leaky_relu(0.2)
    ebuf[t] = x;
    atomicMax(&emax[d * HEADS + h], fkey(x));
}

// Pass B: exp(score - max), store, atomic segment sum of denominators
template<int HEADS>
__global__ void edge_softmax(const int* __restrict__ ei,
                             const unsigned* __restrict__ emax,
                             float* __restrict__ ebuf, float* __restrict__ denom)
{
    int t = blockIdx.x * blockDim.x + threadIdx.x;
    if (t >= EPRIME * HEADS) return;
    const int e = t / HEADS, h = t % HEADS;
    const int d = edst(ei, e);
    const float m = funkey(emax[d * HEADS + h]);
    const float ex = __expf(ebuf[t] - m);
    ebuf[t] = ex;
    atomicAdd(&denom[d * HEADS + h], ex);
}

// Pass C: acc[dst] += alpha * H[src].  One wave per edge; lane covers
// F/32 contiguous features (8 for 4 heads, 2 for 1 head); since HID=64
// each lane's chunk lies inside a single head.
template<int HEADS>
__global__ void edge_aggregate(const int* __restrict__ ei,
                               const float* __restrict__ Hf,
                               const float* __restrict__ ebuf,
                               const float* __restrict__ denom,
                               float* __restrict__ acc)
{
    constexpr int F   = HEADS * HID;
    constexpr int PER = F / 32;
    const int e    = (int)((blockIdx.x * (long long)blockDim.x + threadIdx.x) >> 5);
    const int lane = threadIdx.x & 31;
    if (e >= EPRIME) return;
    const int s = esrc(ei, e), d = edst(ei, e);
    const int off = lane * PER;
    const int h   = off / HID;
    const float alpha = ebuf[e * HEADS + h] / (denom[d * HEADS + h] + 1e-16f);
    const float* hs = Hf  + (size_t)s * F + off;
    float*       ap = acc + (size_t)d * F + off;
    #pragma unroll
    for (int j = 0; j < PER; ++j) atomicAdd(&ap[j], alpha * hs[j]);
}

// acc = elu(acc + bias) in place
template<int F>
__global__ void bias_elu(float* __restrict__ acc, const float* __restrict__ bias)
{
    int t = blockIdx.x * blockDim.x + threadIdx.x;
    if (t >= NNODES * F) return;
    const float v = acc[t] + bias[t % F];
    acc[t] = (v > 0.f) ? v : expm1f(v);
}

// global mean pool: atomic segment sum + counts
__global__ void pool_accum(const int* __restrict__ batch, const float* __restrict__ h2,
                           float* __restrict__ pools, float* __restrict__ cnt)
{
    int t = blockIdx.x * blockDim.x + threadIdx.x;
    if (t >= NNODES * HID) return;
    const int n = t / HID, c = t % HID;
    const int g = batch[n];
    atomicAdd(&pools[g * HID + c], h2[t]);
    if (c == 0) atomicAdd(&cnt[g], 1.0f);
}

__global__ void pool_final(const float* __restrict__ pools, const float* __restrict__ cnt,
                           float* __restrict__ out)
{
    int t = blockIdx.x * blockDim.x + threadIdx.x;
    if (t >= NGRAPH * HID) return;
    out[t] = pools[t] / fmaxf(cnt[t / HID], 1.0f);
}

// =====================================================================
// Host-side pipeline
// =====================================================================
namespace {

// workspace layout (float offsets); total ~119 MB
constexpr size_t OFF_H1   = 0;                                   // 12.8M (layer1 feats;
                                                                 // layer2 reuses: [0,3.2M)=H2, [3.2M,6.4M)=ACC2)
constexpr size_t OFF_ACC1 = OFF_H1   + (size_t)NNODES * F1;      // 12.8M
constexpr size_t OFF_EBUF = OFF_ACC1 + (size_t)NNODES * F1;      // 3.4M
constexpr size_t OFF_AS   = OFF_EBUF + (size_t)EPRIME * HEADS1;  // 200k
constexpr size_t OFF_AD   = OFF_AS   + (size_t)NNODES * HEADS1;
constexpr size_t OFF_EMAX = OFF_AD   + (size_t)NNODES * HEADS1;
constexpr size_t OFF_DEN  = OFF_EMAX + (size_t)NNODES * HEADS1;
constexpr size_t OFF_POOL = OFF_DEN  + (size_t)NNODES * HEADS1;
constexpr size_t OFF_CNT  = OFF_POOL + (size_t)NGRAPH * HID;

inline int cdiv(long long a, int b) { return (int)((a + b - 1) / b); }

void run_graph(const float* x, const int* ei, const int* batch,
               const float* W1, const float* a_src1, const float* a_dst1, const float* b1,
               const float* W2, const float* a_src2, const float* a_dst2, const float* b2,
               float* ws, float* out, hipStream_t stream)
{
    float*    H1   = ws + OFF_H1;
    float*    ACC1 = ws + OFF_ACC1;
    float*    H2   = ws + OFF_H1;                      // reuse after layer1
    float*    ACC2 = ws + OFF_H1 + (size_t)NNODES * HID;
    float*    EB   = ws + OFF_EBUF;
    float*    AS   = ws + OFF_AS;
    float*    AD   = ws + OFF_AD;
    unsigned* EMAX = (unsigned*)(ws + OFF_EMAX);
    float*    DEN  = ws + OFF_DEN;
    float*    POOL = ws + OFF_POOL;
    float*    CNT  = ws + OFF_CNT;

    const int mBlocks = cdiv(NNODES, 256);             // 196

    // ---------------- layer 1 (heads=4) ----------------
    hipMemsetAsync(ACC1, 0, (size_t)NNODES * F1 * sizeof(float), stream);
    hipMemsetAsync(EMAX, 0, (size_t)NNODES * HEADS1 * sizeof(unsigned), stream);
    hipMemsetAsync(DEN,  0, (size_t)NNODES * HEADS1 * sizeof(float), stream);
    hipMemsetAsync(POOL, 0, ((size_t)NGRAPH * HID + NGRAPH) * sizeof(float), stream);

    gemm_wmma_lds<INDIM, F1><<<dim3(mBlocks, F1 / 64), 256, 0, stream>>>(x, W1, H1);
    attn_coef<HEADS1><<<cdiv((long long)NNODES * HEADS1, 256), 256, 0, stream>>>(
        H1, a_src1, a_dst1, AS, AD);
    edge_scores<HEADS1><<<cdiv((long long)EPRIME * HEADS1, 256), 256, 0, stream>>>(
        ei, AS, AD, EB, EMAX);
    edge_softmax<HEADS1><<<cdiv((long long)EPRIME * HEADS1, 256), 256, 0, stream>>>(
        ei, EMAX, EB, DEN);
    edge_aggregate<HEADS1><<<cdiv((long long)EPRIME * 32, 256), 256, 0, stream>>>(
        ei, H1, EB, DEN, ACC1);
    bias_elu<F1><<<cdiv((long long)NNODES * F1, 256), 256, 0, stream>>>(ACC1, b1);

    // ---------------- layer 2 (heads=1) ----------------
    hipMemsetAsync(ACC2, 0, (size_t)NNODES * HID * sizeof(float), stream);
    hipMemsetAsync(EMAX, 0, (size_t)NNODES * sizeof(unsigned), stream);
    hipMemsetAsync(DEN,  0, (size_t)NNODES * sizeof(float), stream);

    gemm_wmma_lds<F1, HID><<<dim3(mBlocks, 1), 256, 0, stream>>>(ACC1, W2, H2);
    attn_coef<1><<<cdiv(NNODES, 256), 256, 0, stream>>>(H2, a_src2, a_dst2, AS, AD);
    edge_scores<1><<<cdiv(EPRIME, 256), 256, 0, stream>>>(ei, AS, AD, EB, EMAX);
    edge_softmax<1><<<cdiv(EPRIME, 256), 256, 0, stream>>>(ei, EMAX, EB, DEN);
    edge_aggregate<1><<<cdiv((long long)EPRIME * 32, 256), 256, 0, stream>>>(
        ei, H2, EB, DEN, ACC2);
    bias_elu<HID><<<cdiv((long long)NNODES * HID, 256), 256, 0, stream>>>(ACC2, b2);

    // ---------------- mean pool ----------------
    pool_accum<<<cdiv((long long)NNODES * HID, 256), 256, 0, stream>>>(batch, ACC2, POOL, CNT);
    pool_final<<<cdiv(NGRAPH * HID, 256), 256, 0, stream>>>(POOL, CNT, out);
}

} // namespace

extern "C" void kernel_launch(void* const* d_in, const int* in_sizes, int n_in,
                              void* d_out, int out_size, void* d_ws, size_t ws_size,
                              hipStream_t stream)
{
    // setup_inputs order:
    // 0:x1 1:edge_index1 2:batch1 3:x2 4:edge_index2 5:batch2
    // 6:W1 7:a_src1 8:a_dst1 9:b1 10:W2 11:a_src2 12:a_dst2 13:b2
    const float* x1  = (const float*)d_in[0];
    const int*   ei1 = (const int*)  d_in[1];
    const int*   bt1 = (const int*)  d_in[2];
    const float* x2  = (const float*)d_in[3];
    const int*   ei2 = (const int*)  d_in[4];
    const int*   bt2 = (const int*)  d_in[5];
    const float* W1  = (const float*)d_in[6];
    const float* as1 = (const float*)d_in[7];
    const float* ad1 = (const float*)d_in[8];
    const float* b1  = (const float*)d_in[9];
    const float* W2  = (const float*)d_in[10];
    const float* as2 = (const float*)d_in[11];
    const float* ad2 = (const float*)d_in[12];
    const float* b2  = (const float*)d_in[13];

    float* ws  = (float*)d_ws;
    float* out = (float*)d_out;   // [emb1 (64x64) | emb2 (64x64)]

    run_graph(x1, ei1, bt1, W1, as1, ad1, b1, W2, as2, ad2, b2,
              ws, out, stream);
    run_graph(x2, ei2, bt2, W1, as1, ad1, b1, W2, as2, ad2, b2,
              ws, out + (size_t)NGRAPH * HID, stream);
}